// Model_23252952940854
// MI455X (gfx1250) — compile-verified
//
#include <hip/hip_runtime.h>

#define NN 100000
#define D 128
#define ND ((size_t)NN * D)
#define LDSP 132   // padded LDS row stride (floats): bank stride 4 -> conflict-free fragment reads

typedef __attribute__((ext_vector_type(2))) float v2f;
typedef __attribute__((ext_vector_type(8))) float v8f;

__device__ __forceinline__ void atomAddF(float* p, float v) {
    __hip_atomic_fetch_add(p, v, __ATOMIC_RELAXED, __HIP_MEMORY_SCOPE_AGENT);
}

// Accumulate two 16x16 fp32 tiles (rows [0:16) and [16:32) of the staged A block):
//   acc[mt] += As[mt*16:+16, 0:128] @ W[0:128, col0:+16]
// B fragments are loaded once per k-step and reused by both M-tiles.
// WMMA f32 16x16x4 VGPR layouts (ISA 7.12.2):
//   A 16x4:  lanes 0-15 -> M=lane, v0=K0,v1=K1 ; lanes 16-31 -> M=lane-16, v0=K2,v1=K3
//   B 4x16:  lanes 0-15 -> N=lane, v0=K0,v1=K1 ; lanes 16-31 -> N=lane-16, v0=K2,v1=K3
//   C/D:     VGPR r: lanes 0-15 -> M=r ; lanes 16-31 -> M=r+8
__device__ __forceinline__ void wmma_pass(const float (*__restrict__ As)[LDSP],
                                          const float* __restrict__ W,
                                          int col0, int half, int l,
                                          v8f& acc0, v8f& acc1) {
    const float* wcol = W + col0 + l;
#pragma unroll
    for (int kk = 0; kk < D; kk += 4) {
        const int k0 = kk + 2 * half;
        v2f b;
        b.x = wcol[(size_t)k0 * D];
        b.y = wcol[(size_t)(k0 + 1) * D];
        v2f a0, a1;
        a0.x = As[l][k0];        a0.y = As[l][k0 + 1];        // ds_load_b64
        a1.x = As[16 + l][k0];   a1.y = As[16 + l][k0 + 1];
        acc0 = __builtin_amdgcn_wmma_f32_16x16x4_f32(false, a0, false, b, (short)0, acc0, false, false);
        acc1 = __builtin_amdgcn_wmma_f32_16x16x4_f32(false, a1, false, b, (short)0, acc1, false, false);
    }
}

// C[M,128] = act( A0 @ W0 (+ A1 @ W1) + bias )
// One block = 32 rows x 128 cols: 8 waves, each wave owns a 16-col slice and two 16-row tiles.
template <bool RELU, bool TWO>
__global__ __launch_bounds__(256) void gemm_sage(const float* __restrict__ A0,
                                                 const float* __restrict__ W0,
                                                 const float* __restrict__ A1,
                                                 const float* __restrict__ W1,
                                                 const float* __restrict__ bias,
                                                 float* __restrict__ C) {
    __shared__ float As0[32][LDSP];
    __shared__ float As1[TWO ? 32 : 1][LDSP];

    const int row0 = blockIdx.x * 32;
    const int col0 = (threadIdx.x >> 5) * 16;
    const int lane = threadIdx.x & 31;
    const int half = lane >> 4;
    const int l = lane & 15;

    // Cooperative stage: 256 threads x (32 rows x 128 cols): each thread copies 16 floats.
    {
        const int r = threadIdx.x >> 3;          // 0..31
        const int c = (threadIdx.x & 7) * 16;    // 0,16,...,112
        const float* g0 = A0 + (size_t)(row0 + r) * D + c;
#pragma unroll
        for (int j = 0; j < 16; j += 4)
            *(float4*)&As0[r][c + j] = *(const float4*)(g0 + j);
        if (TWO) {
            const float* g1 = A1 + (size_t)(row0 + r) * D + c;
#pragma unroll
            for (int j = 0; j < 16; j += 4)
                *(float4*)&As1[r][c + j] = *(const float4*)(g1 + j);
        }
    }
    __syncthreads();

    v8f acc0 = {0.f, 0.f, 0.f, 0.f, 0.f, 0.f, 0.f, 0.f};
    v8f acc1 = acc0;
    wmma_pass(As0, W0, col0, half, l, acc0, acc1);
    if (TWO) wmma_pass(As1, W1, col0, half, l, acc0, acc1);

    const float bc = bias[col0 + l];
#pragma unroll
    for (int r = 0; r < 8; ++r) {
        float v0 = acc0[r] + bc;
        float v1 = acc1[r] + bc;
        if (RELU) { v0 = fmaxf(v0, 0.f); v1 = fmaxf(v1, 0.f); }
        C[(size_t)(row0 + half * 8 + r) * D + col0 + l] = v0;
        C[(size_t)(row0 + 16 + half * 8 + r) * D + col0 + l] = v1;
    }
}

__global__ __launch_bounds__(256) void zero_f32(float* __restrict__ p, long n) {
    long i = (long)blockIdx.x * blockDim.x + threadIdx.x;
    if (i < n) p[i] = 0.f;
}

// One wave32 per edge: agg[dst] += h[src]; deg[dst] += 1
__global__ __launch_bounds__(256) void sage_scatter(const float* __restrict__ h,
                                                    const int* __restrict__ src,
                                                    const int* __restrict__ dst,
                                                    float* __restrict__ agg,
                                                    float* __restrict__ deg, int E) {
    long t = (long)blockIdx.x * blockDim.x + threadIdx.x;
    int e = (int)(t >> 5);
    if (e >= E) return;
    int lane = (int)(t & 31);
    int s = src[e], d = dst[e];
    const float4 v = *(const float4*)(h + (size_t)s * D + lane * 4);
    float* p = agg + (size_t)d * D + lane * 4;
    atomAddF(p + 0, v.x);
    atomAddF(p + 1, v.y);
    atomAddF(p + 2, v.z);
    atomAddF(p + 3, v.w);
    if (lane == 0) atomAddF(deg + d, 1.0f);
}

__global__ __launch_bounds__(256) void sage_mean(float* __restrict__ agg,
                                                 const float* __restrict__ deg) {
    long t = (long)blockIdx.x * blockDim.x + threadIdx.x;
    int n = (int)(t >> 5);
    if (n >= NN) return;
    int lane = (int)(t & 31);
    float inv = 1.0f / fmaxf(deg[n], 1.0f);
    float4* p = (float4*)(agg + (size_t)n * D + lane * 4);
    float4 v = *p;
    v.x *= inv; v.y *= inv; v.z *= inv; v.w *= inv;
    *p = v;
}

// One wave32 per edge pair: out[e] = dot(h[src[e]], h[dst[e]])
__global__ __launch_bounds__(256) void edge_dot(const float* __restrict__ h,
                                                const int* __restrict__ src,
                                                const int* __restrict__ dst,
                                                float* __restrict__ out, int n) {
    long t = (long)blockIdx.x * blockDim.x + threadIdx.x;
    int e = (int)(t >> 5);
    if (e >= n) return;
    int lane = (int)(t & 31);
    const float4 a = *(const float4*)(h + (size_t)src[e] * D + lane * 4);
    const float4 b = *(const float4*)(h + (size_t)dst[e] * D + lane * 4);
    float s = a.x * b.x + a.y * b.y + a.z * b.z + a.w * b.w;
#pragma unroll
    for (int off = 16; off; off >>= 1) s += __shfl_down(s, off, 32);
    if (lane == 0) out[e] = s;
}

extern "C" void kernel_launch(void* const* d_in, const int* in_sizes, int n_in,
                              void* d_out, int out_size, void* d_ws, size_t ws_size,
                              hipStream_t stream) {
    const float* feat    = (const float*)d_in[0];
    const int*   e0s     = (const int*)d_in[1];
    const int*   e0d     = (const int*)d_in[2];
    const int*   e1s     = (const int*)d_in[3];
    const int*   e1d     = (const int*)d_in[4];
    const int*   ps      = (const int*)d_in[5];
    const int*   pd      = (const int*)d_in[6];
    const int*   ns      = (const int*)d_in[7];
    const int*   nd      = (const int*)d_in[8];
    const float* w_proj  = (const float*)d_in[9];
    const float* b_proj  = (const float*)d_in[10];
    const float* w_self1 = (const float*)d_in[11];
    const float* w_neigh1= (const float*)d_in[12];
    const float* b1      = (const float*)d_in[13];
    const float* w_self2 = (const float*)d_in[14];
    const float* w_neigh2= (const float*)d_in[15];
    const float* b2      = (const float*)d_in[16];

    const int E0   = in_sizes[1];
    const int E1   = in_sizes[3];
    const int nPos = in_sizes[5];
    const int nNeg = in_sizes[7];

    // Workspace: h0 | agg | deg | h1   (h2 reuses h0's buffer)
    float* h0  = (float*)d_ws;
    float* agg = h0 + ND;
    float* deg = agg + ND;
    float* h1  = deg + NN;
    float* h2  = h0;
    float* out = (float*)d_out;

    const int mtiles = NN / 32;                       // 3125 (exact: 100000 % 32 == 0)
    const long zn    = (long)ND + NN;                 // agg + deg are contiguous
    const int zgrid  = (int)((zn + 255) / 256);
    const int sgrid0 = (int)(((long)E0 * 32 + 255) / 256);
    const int sgrid1 = (int)(((long)E1 * 32 + 255) / 256);
    const int mgrid  = (int)(((long)NN * 32 + 255) / 256);

    // h0 = relu(feat @ Wp + bp)
    gemm_sage<true, false><<<mtiles, 256, 0, stream>>>(feat, w_proj, nullptr, nullptr, b_proj, h0);

    // Layer 1: mean aggregation over edge0, then h1 = relu(h0@Ws1 + mean@Wn1 + b1)
    zero_f32<<<zgrid, 256, 0, stream>>>(agg, zn);
    sage_scatter<<<sgrid0, 256, 0, stream>>>(h0, e0s, e0d, agg, deg, E0);
    sage_mean<<<mgrid, 256, 0, stream>>>(agg, deg);
    gemm_sage<true, true><<<mtiles, 256, 0, stream>>>(h0, w_self1, agg, w_neigh1, b1, h1);

    // Layer 2: mean aggregation over edge1, then h2 = h1@Ws2 + mean@Wn2 + b2 (no relu)
    zero_f32<<<zgrid, 256, 0, stream>>>(agg, zn);
    sage_scatter<<<sgrid1, 256, 0, stream>>>(h1, e1s, e1d, agg, deg, E1);
    sage_mean<<<mgrid, 256, 0, stream>>>(agg, deg);
    gemm_sage<false, true><<<mtiles, 256, 0, stream>>>(h1, w_self2, agg, w_neigh2, b2, h2);

    // Edge scoring
    edge_dot<<<(int)(((long)nPos * 32 + 255) / 256), 256, 0, stream>>>(h2, ps, pd, out, nPos);
    edge_dot<<<(int)(((long)nNeg * 32 + 255) / 256), 256, 0, stream>>>(h2, ns, nd, out + nPos, nNeg);
}